// CTAN_37434934952569
// MI455X (gfx1250) — compile-verified
//
#include <hip/hip_runtime.h>
#include <climits>

// ---------------------------------------------------------------------------
// CTAN temporal-graph attention step for MI455X (gfx1250, wave32, WMMA).
// All dense GEMMs use V_WMMA_F32_16X16X4_F32 (full f32 precision on the
// matrix pipe). Weights staged in LDS in WMMA K-pair layout.
// ---------------------------------------------------------------------------

typedef __attribute__((ext_vector_type(2))) float v2f;
typedef __attribute__((ext_vector_type(8))) float v8f;

#define MEMC 128
#define TIMEC 64
#define EDGEC 64
#define EPSILONC 0.1f
#define GAMMAC 0.1f
#define SCALEC 0.08838834764831845f  // 1/sqrt(128)

__device__ __forceinline__ v8f wmma4(v2f a, v2f b, v8f c) {
  // D = A(16x4 f32) * B(4x16 f32) + C(16x16 f32)
  return __builtin_amdgcn_wmma_f32_16x16x4_f32(
      /*neg_a=*/false, a, /*neg_b=*/false, b,
      /*c_mod=*/(short)0, c, /*reuse_a=*/false, /*reuse_b=*/false);
}

// Monotonic int transform for float atomic-max
__device__ __forceinline__ int f2ord(float x) {
  int f = __float_as_int(x);
  return (f >= 0) ? f : (f ^ 0x7FFFFFFF);
}
__device__ __forceinline__ float ord2f(int m) {
  int bits = (m >= 0) ? m : (m ^ 0x7FFFFFFF);
  return __int_as_float(bits);
}

// ---------------------------------------------------------------------------
// Init: zero aggregation buffer (d_out), set amax sentinel, zero denom
// ---------------------------------------------------------------------------
__global__ __launch_bounds__(256) void init_kernel(int* amax_i, float* denom,
                                                   float* agg, int n_nodes) {
  int i = blockIdx.x * 256 + threadIdx.x;
  if (i < n_nodes * MEMC) agg[i] = 0.0f;
  if (i < n_nodes) { amax_i[i] = INT_MIN; denom[i] = 0.0f; }
}

// ---------------------------------------------------------------------------
// A_anti = W_anti - W_anti^T - GAMMA*I
// ---------------------------------------------------------------------------
__global__ __launch_bounds__(256) void prep_anti_kernel(const float* __restrict__ Wanti,
                                                        float* __restrict__ Aanti) {
  int i = blockIdx.x * 256 + threadIdx.x;
  if (i < MEMC * MEMC) {
    int r = i >> 7, c = i & 127;
    Aanti[i] = Wanti[r * MEMC + c] - Wanti[c * MEMC + r] - ((r == c) ? GAMMAC : 0.0f);
  }
}

// ---------------------------------------------------------------------------
// Edge kernel: edge_attr = [msg | cos(rel_t*w_time+b_time)], eproj = attr@We.T
// 128 edges per block (8 waves, one 16-row tile each), WMMA f32 GEMM K=128.
// ---------------------------------------------------------------------------
#define AP_STRIDE 65
__global__ __launch_bounds__(256) void edge_proj_kernel(
    const float* __restrict__ lastup, const int* __restrict__ eidx,
    const float* __restrict__ t, const float* __restrict__ msg,
    const float* __restrict__ w_time, const float* __restrict__ b_time,
    const float* __restrict__ We, float* __restrict__ eproj, int E_) {
  __shared__ v2f Ap[128 * AP_STRIDE];  // edge_attr, K-pair layout, padded
  __shared__ v2f Wp[64 * 128];         // We^T pairs: Wp[j][n] = {We[n][2j], We[n][2j+1]}
  __shared__ float relt[128];

  const int tid = threadIdx.x;
  const int ebase = blockIdx.x * 128;

  // stage weights (coalesced float2 reads)
  for (int i = tid; i < 64 * 128; i += 256) {
    int n = i >> 6, j = i & 63;
    float2 w = *(const float2*)(We + (size_t)n * MEMC + 2 * j);
    v2f v; v.x = w.x; v.y = w.y;
    Wp[j * 128 + n] = v;
  }
  // per-edge relative time
  if (tid < 128) {
    int e = ebase + tid;
    float r = 0.0f;
    if (e < E_) {
      int s = eidx[e];
      r = fabsf(lastup[s] - t[e]);  // (|dt| - MEAN_DT)/STD_DT with 0/1
    }
    relt[tid] = r;
  }
  __syncthreads();

  // stage edge_attr pairs
  for (int i = tid; i < 128 * 64; i += 256) {
    int row = i >> 6, j = i & 63;
    int e = ebase + row;
    v2f val; val.x = 0.0f; val.y = 0.0f;
    if (e < E_) {
      int k = 2 * j;
      if (k < EDGEC) {
        float2 m = *(const float2*)(msg + (size_t)e * EDGEC + k);
        val.x = m.x; val.y = m.y;
      } else {
        float r = relt[row];
        int k0 = k - EDGEC;
        val.x = cosf(r * w_time[k0]     + b_time[k0]);
        val.y = cosf(r * w_time[k0 + 1] + b_time[k0 + 1]);
      }
    }
    Ap[row * AP_STRIDE + j] = val;
  }
  __syncthreads();

  // WMMA GEMM: each wave -> one 16-row tile x 8 col-tiles
  const int wave = tid >> 5, lane = tid & 31;
  const int half = lane >> 4, lcol = lane & 15;
  const int rowloc = wave * 16 + lcol;

  v8f acc[8] = {};
  for (int kk = 0; kk < 128; kk += 4) {
    int jj = (kk >> 1) + half;
    v2f a = Ap[rowloc * AP_STRIDE + jj];
#pragma unroll
    for (int n = 0; n < 8; ++n) {
      v2f b = Wp[jj * 128 + n * 16 + lcol];
      acc[n] = wmma4(a, b, acc[n]);
    }
  }
  // store D: VGPR r -> (M = r + 8*half, N = lcol) within tile
  const int erow0 = ebase + wave * 16 + half * 8;
#pragma unroll
  for (int n = 0; n < 8; ++n) {
    int col = n * 16 + lcol;
#pragma unroll
    for (int r = 0; r < 8; ++r) {
      int e = erow0 + r;
      if (e < E_) eproj[(size_t)e * MEMC + col] = acc[n][r];
    }
  }
}

// ---------------------------------------------------------------------------
// h = x @ W_enc.T + b_enc   (K = 256, weights staged in two 64KB halves)
// ---------------------------------------------------------------------------
__global__ __launch_bounds__(256) void node_h_kernel(
    const float* __restrict__ x, const float* __restrict__ W,
    const float* __restrict__ bias, float* __restrict__ h, int rows) {
  __shared__ v2f Wp[64 * 128];
  const int tid = threadIdx.x, wave = tid >> 5, lane = tid & 31;
  const int half = lane >> 4, lcol = lane & 15;
  const int rowbase = blockIdx.x * 128 + wave * 16;
  const bool active = rowbase < rows;
  const int aRow = active ? min(rowbase + lcol, rows - 1) : 0;

  v8f acc[8] = {};
  for (int kh = 0; kh < 2; ++kh) {
    __syncthreads();
    for (int i = tid; i < 64 * 128; i += 256) {
      int n = i >> 6, j = i & 63;
      float2 w = *(const float2*)(W + (size_t)n * 256 + kh * 128 + 2 * j);
      v2f v; v.x = w.x; v.y = w.y;
      Wp[j * 128 + n] = v;
    }
    __syncthreads();
    if (active) {
      const float* arow = x + (size_t)aRow * 256 + kh * 128;
      for (int kk = 0; kk < 128; kk += 4) {
        int jj = (kk >> 1) + half;
        v2f a = *(const v2f*)(arow + kk + 2 * half);
#pragma unroll
        for (int n = 0; n < 8; ++n) {
          v2f b = Wp[jj * 128 + n * 16 + lcol];
          acc[n] = wmma4(a, b, acc[n]);
        }
      }
    }
  }
  if (active) {
    int r0 = rowbase + half * 8;
#pragma unroll
    for (int n = 0; n < 8; ++n) {
      int col = n * 16 + lcol;
      float bv = bias[col];
#pragma unroll
      for (int r = 0; r < 8; ++r)
        if (r0 + r < rows) h[(size_t)(r0 + r) * MEMC + col] = acc[n][r] + bv;
    }
  }
}

// ---------------------------------------------------------------------------
// Q/K/V/AH = h @ {Wq,Wk,Wv,Aanti}.T + {bq,bk,bv,0}  (K=128, 4 weights staged
// sequentially in one 64KB LDS buffer)
// ---------------------------------------------------------------------------
__global__ __launch_bounds__(256) void node_qkv_kernel(
    const float* __restrict__ h,
    const float* Wq, const float* Wk, const float* Wv, const float* Wa,
    const float* bq, const float* bk, const float* bv,
    float* Q, float* K, float* V, float* AH, int rows) {
  __shared__ v2f Wp[64 * 128];
  const int tid = threadIdx.x, wave = tid >> 5, lane = tid & 31;
  const int half = lane >> 4, lcol = lane & 15;
  const int rowbase = blockIdx.x * 128 + wave * 16;
  const bool active = rowbase < rows;
  const int aRow = active ? min(rowbase + lcol, rows - 1) : 0;

  const float* Ws[4] = {Wq, Wk, Wv, Wa};
  const float* Bs[4] = {bq, bk, bv, nullptr};
  float* Os[4] = {Q, K, V, AH};

  for (int m = 0; m < 4; ++m) {
    __syncthreads();
    const float* W = Ws[m];
    for (int i = tid; i < 64 * 128; i += 256) {
      int n = i >> 6, j = i & 63;
      float2 w = *(const float2*)(W + (size_t)n * MEMC + 2 * j);
      v2f v; v.x = w.x; v.y = w.y;
      Wp[j * 128 + n] = v;
    }
    __syncthreads();
    if (active) {
      v8f acc[8] = {};
      const float* arow = h + (size_t)aRow * MEMC;
      for (int kk = 0; kk < 128; kk += 4) {
        int jj = (kk >> 1) + half;
        v2f a = *(const v2f*)(arow + kk + 2 * half);
#pragma unroll
        for (int n = 0; n < 8; ++n) {
          v2f b = Wp[jj * 128 + n * 16 + lcol];
          acc[n] = wmma4(a, b, acc[n]);
        }
      }
      int r0 = rowbase + half * 8;
      float* O = Os[m];
      const float* bias = Bs[m];
#pragma unroll
      for (int n = 0; n < 8; ++n) {
        int col = n * 16 + lcol;
        float bvv = bias ? bias[col] : 0.0f;
#pragma unroll
        for (int r = 0; r < 8; ++r)
          if (r0 + r < rows) O[(size_t)(r0 + r) * MEMC + col] = acc[n][r] + bvv;
      }
    }
  }
}

// ---------------------------------------------------------------------------
// Pass 1: alpha[e] = scale * dot(Q[dst], K[src]+eproj[e]); atomic max per dst
// One wave (32 lanes x float4) per edge.
// ---------------------------------------------------------------------------
__global__ __launch_bounds__(256) void alpha_kernel(
    const int* __restrict__ eidx, const float* __restrict__ Q,
    const float* __restrict__ Kf, const float* __restrict__ eproj,
    float* __restrict__ alpha, int* __restrict__ amax_i, int E_) {
  int gw = (blockIdx.x * 256 + threadIdx.x) >> 5;
  int lane = threadIdx.x & 31;
  if (gw >= E_) return;
  int e = gw;
  int s = eidx[e], d = eidx[E_ + e];
  float4 q  = *(const float4*)(Q     + (size_t)d * MEMC + lane * 4);
  float4 kv = *(const float4*)(Kf    + (size_t)s * MEMC + lane * 4);
  float4 ep = *(const float4*)(eproj + (size_t)e * MEMC + lane * 4);
  float p = q.x * (kv.x + ep.x) + q.y * (kv.y + ep.y) +
            q.z * (kv.z + ep.z) + q.w * (kv.w + ep.w);
#pragma unroll
  for (int off = 16; off > 0; off >>= 1) p += __shfl_xor(p, off, 32);
  if (lane == 0) {
    float a = p * SCALEC;
    alpha[e] = a;
    atomicMax(amax_i + d, f2ord(a));
  }
}

// ---------------------------------------------------------------------------
// Pass 2: denom[dst] += exp(alpha - amax[dst])
// ---------------------------------------------------------------------------
__global__ __launch_bounds__(256) void denom_kernel(
    const int* __restrict__ eidx, const float* __restrict__ alpha,
    const int* __restrict__ amax_i, float* __restrict__ denom, int E_) {
  int e = blockIdx.x * 256 + threadIdx.x;
  if (e >= E_) return;
  int d = eidx[E_ + e];
  atomicAdd(denom + d, __expf(alpha[e] - ord2f(amax_i[d])) * 0.0f + expf(alpha[e] - ord2f(amax_i[d])));
}

// ---------------------------------------------------------------------------
// Pass 3: agg[dst] += w * (V[src] + eproj[e]),  w = exp(alpha-amax)/(denom+eps)
// One wave per edge, 4 cols per lane, f32 global atomics.
// ---------------------------------------------------------------------------
__global__ __launch_bounds__(256) void agg_kernel(
    const int* __restrict__ eidx, const float* __restrict__ alpha,
    const int* __restrict__ amax_i, const float* __restrict__ denom,
    const float* __restrict__ Vf, const float* __restrict__ eproj,
    float* __restrict__ agg, int E_) {
  int gw = (blockIdx.x * 256 + threadIdx.x) >> 5;
  int lane = threadIdx.x & 31;
  if (gw >= E_) return;
  int e = gw;
  int s = eidx[e], d = eidx[E_ + e];
  float w = expf(alpha[e] - ord2f(amax_i[d])) / (denom[d] + 1e-16f);
  float4 v  = *(const float4*)(Vf    + (size_t)s * MEMC + lane * 4);
  float4 ep = *(const float4*)(eproj + (size_t)e * MEMC + lane * 4);
  float* o = agg + (size_t)d * MEMC + lane * 4;
  atomicAdd(o + 0, w * (v.x + ep.x));
  atomicAdd(o + 1, w * (v.y + ep.y));
  atomicAdd(o + 2, w * (v.z + ep.z));
  atomicAdd(o + 3, w * (v.w + ep.w));
}

// ---------------------------------------------------------------------------
// Final: g = tanh(AH + agg + b_anti); out = tanh(h + eps*g)   (in-place on agg)
// ---------------------------------------------------------------------------
__global__ __launch_bounds__(256) void final_kernel(
    const float* __restrict__ h, const float* __restrict__ AH,
    const float* __restrict__ b_anti, float* __restrict__ out, int total) {
  int i = blockIdx.x * 256 + threadIdx.x;
  if (i >= total) return;
  float g = tanhf(AH[i] + out[i] + b_anti[i & 127]);
  out[i] = tanhf(h[i] + EPSILONC * g);
}

// ---------------------------------------------------------------------------
extern "C" void kernel_launch(void* const* d_in, const int* in_sizes, int n_in,
                              void* d_out, int out_size, void* d_ws, size_t ws_size,
                              hipStream_t stream) {
  const float* x        = (const float*)d_in[0];
  const float* last_up  = (const float*)d_in[1];
  const int*   eidx     = (const int*)  d_in[2];
  const float* t        = (const float*)d_in[3];
  const float* msg      = (const float*)d_in[4];
  const float* w_time   = (const float*)d_in[5];
  const float* b_time   = (const float*)d_in[6];
  const float* W_enc    = (const float*)d_in[7];
  const float* b_enc    = (const float*)d_in[8];
  const float* Wq       = (const float*)d_in[9];
  const float* bq       = (const float*)d_in[10];
  const float* Wk       = (const float*)d_in[11];
  const float* bk       = (const float*)d_in[12];
  const float* Wv       = (const float*)d_in[13];
  const float* bv       = (const float*)d_in[14];
  const float* We       = (const float*)d_in[15];
  const float* W_anti   = (const float*)d_in[16];
  const float* b_anti   = (const float*)d_in[17];

  const int N_ = in_sizes[1];  // last_update: [N]
  const int E_ = in_sizes[3];  // t: [E]
  float* out = (float*)d_out;  // doubles as aggregation accumulator

  // Workspace carve (~515 MB for N=1e5, E=5e5)
  char* wp = (char*)d_ws;
  auto carve = [&](size_t bytes) -> char* {
    char* p = wp;
    wp += (bytes + 255) & ~(size_t)255;
    return p;
  };
  float* h      = (float*)carve((size_t)N_ * MEMC * 4);
  float* Qb     = (float*)carve((size_t)N_ * MEMC * 4);
  float* Kb     = (float*)carve((size_t)N_ * MEMC * 4);
  float* Vb     = (float*)carve((size_t)N_ * MEMC * 4);
  float* AH     = (float*)carve((size_t)N_ * MEMC * 4);
  float* eproj  = (float*)carve((size_t)E_ * MEMC * 4);
  float* alpha  = (float*)carve((size_t)E_ * 4);
  int*   amax_i = (int*)  carve((size_t)N_ * 4);
  float* denom  = (float*)carve((size_t)N_ * 4);
  float* Aanti  = (float*)carve((size_t)MEMC * MEMC * 4);

  const int nm = N_ * MEMC;
  init_kernel<<<(nm + 255) / 256, 256, 0, stream>>>(amax_i, denom, out, N_);
  prep_anti_kernel<<<(MEMC * MEMC + 255) / 256, 256, 0, stream>>>(W_anti, Aanti);

  edge_proj_kernel<<<(E_ + 127) / 128, 256, 0, stream>>>(
      last_up, eidx, t, msg, w_time, b_time, We, eproj, E_);

  node_h_kernel<<<(N_ + 127) / 128, 256, 0, stream>>>(x, W_enc, b_enc, h, N_);

  node_qkv_kernel<<<(N_ + 127) / 128, 256, 0, stream>>>(
      h, Wq, Wk, Wv, Aanti, bq, bk, bv, Qb, Kb, Vb, AH, N_);

  alpha_kernel<<<(E_ * 32 + 255) / 256, 256, 0, stream>>>(
      eidx, Qb, Kb, eproj, alpha, amax_i, E_);

  denom_kernel<<<(E_ + 255) / 256, 256, 0, stream>>>(eidx, alpha, amax_i, denom, E_);

  agg_kernel<<<(E_ * 32 + 255) / 256, 256, 0, stream>>>(
      eidx, alpha, amax_i, denom, Vb, eproj, out, E_);

  final_kernel<<<(nm + 255) / 256, 256, 0, stream>>>(h, AH, b_anti, out, nm);
}